// ST_Encoder_Module_hyper_27084063768790
// MI455X (gfx1250) — compile-verified
//
#include <hip/hip_runtime.h>

// ---------------------------------------------------------------------------
// ST_Encoder sparse uniform-softmax aggregation for MI455X (gfx1250, wave32)
//   out[:, 0:128]   = (1/deg_ent) * scatter-sum(ent_emb[col] -> row)
//   out[:, 128:256] = (1/deg_rel) * scatter-sum(rel_emb[col] -> row)
// Memory-bound scatter: wave-per-edge coalesced gathers + f32 L2 atomics.
// Final diag-scale pass uses V_WMMA_F32_16X16X4_F32 (D = diag(inv_deg) x S).
// ---------------------------------------------------------------------------

typedef float v2f __attribute__((ext_vector_type(2)));
typedef float v8f __attribute__((ext_vector_type(8)));

#define D_FEAT 128
#define OUT_STRIDE 256

// ---- zero a float buffer (grid-stride) ------------------------------------
__global__ void stenc_zero_f32(float* __restrict__ p, long long n) {
  long long i = (long long)blockIdx.x * blockDim.x + threadIdx.x;
  long long stride = (long long)gridDim.x * blockDim.x;
  for (; i < n; i += stride) p[i] = 0.0f;
}

// ---- degree counting for both graphs in one launch ------------------------
__global__ void stenc_count_deg(const int* __restrict__ entRow, int nEnt,
                                const int* __restrict__ relRow, int nRel,
                                int* __restrict__ degEnt,
                                int* __restrict__ degRel) {
  int i = blockIdx.x * blockDim.x + threadIdx.x;
  if (i < nEnt) {
    atomicAdd(&degEnt[entRow[i]], 1);
  } else {
    int j = i - nEnt;
    if (j < nRel) atomicAdd(&degRel[relRow[j]], 1);
  }
}

// ---- wave-per-edge scatter: 32 lanes x float4 = one 128-float emb row -----
__global__ void stenc_scatter_edges(const int* __restrict__ row,
                                    const int* __restrict__ col,
                                    const float* __restrict__ emb,
                                    float* __restrict__ out,
                                    int nEdges, int colOffset) {
  int tid  = blockIdx.x * blockDim.x + threadIdx.x;
  int wid  = tid >> 5;    // wave id == edge id (uniform per wave)
  int lane = tid & 31;
  if (wid >= nEdges) return;

  int r = row[wid];
  int c = col[wid];

  // coalesced 512B gather of the embedding row (L2-resident: 51MB < 192MB L2)
  const float4 v = ((const float4*)(emb + (size_t)c * D_FEAT))[lane];

  float* dst = out + (size_t)r * OUT_STRIDE + colOffset + lane * 4;
  // hardware global_atomic_add_f32 (no CAS loop)
  unsafeAtomicAdd(dst + 0, v.x);
  unsafeAtomicAdd(dst + 1, v.y);
  unsafeAtomicAdd(dst + 2, v.z);
  unsafeAtomicAdd(dst + 3, v.w);
}

// ---- final scale: D = diag(inv_deg) x S per 16x16 tile via WMMA -----------
// A (16x4 f32, 2 VGPR): v0 = K0 | K2 halves, v1 = K1 | K3 halves (ISA 7.12.2)
// B (4x16 f32, 2 VGPR): v0 = rows k0|k2 striped over lanes, v1 = k1|k3
// C/D (16x16 f32, 8 VGPR): VGPR v -> row v (lanes 0-15), row v+8 (lanes 16-31)
__global__ void stenc_scale_wmma(float* __restrict__ out,
                                 const int* __restrict__ degEnt,
                                 const int* __restrict__ degRel,
                                 int nodeSize, int nTiles) {
  int tid  = blockIdx.x * blockDim.x + threadIdx.x;
  int wid  = tid >> 5;
  int lane = tid & 31;
  if (wid >= nTiles) return;          // uniform per wave

  int rowTile = wid >> 4;             // 256 cols / 16 = 16 col tiles per row
  int colTile = wid & 15;
  int m0 = rowTile * 16;
  int n0 = colTile * 16;
  const int* __restrict__ deg = (n0 < D_FEAT) ? degEnt : degRel;

  int mL  = lane & 15;
  int grp = lane >> 4;

  if (m0 + 16 <= nodeSize) {          // uniform: EXEC stays all-ones for WMMA
    int   d   = deg[m0 + mL];
    float inv = (d > 0) ? (1.0f / (float)d) : 0.0f;

    v8f acc = {};
#pragma unroll
    for (int kc = 0; kc < 4; ++kc) {
      int rlo = 4 * kc + 2 * grp;     // k index of .x element for this half
      v2f b, a;
      b.x = out[(size_t)(m0 + rlo)     * OUT_STRIDE + n0 + mL];
      b.y = out[(size_t)(m0 + rlo + 1) * OUT_STRIDE + n0 + mL];
      a.x = (mL == rlo)     ? inv : 0.0f;   // A[m][k] = inv[m] * (m==k)
      a.y = (mL == rlo + 1) ? inv : 0.0f;
      acc = __builtin_amdgcn_wmma_f32_16x16x4_f32(
          /*neg_a=*/false, a, /*neg_b=*/false, b,
          /*c_mod=*/(short)0, acc, /*reuse_a=*/false, /*reuse_b=*/false);
    }
#pragma unroll
    for (int v = 0; v < 8; ++v)
      out[(size_t)(m0 + v + 8 * grp) * OUT_STRIDE + n0 + mL] = acc[v];
  } else {
    // scalar tail for a partial row-tile (never taken for nodeSize % 16 == 0)
    for (int idx = lane; idx < 256; idx += 32) {
      int r = idx >> 4, cc = idx & 15;
      if (m0 + r < nodeSize) {
        int   d   = deg[m0 + r];
        float inv = (d > 0) ? (1.0f / (float)d) : 0.0f;
        size_t o = (size_t)(m0 + r) * OUT_STRIDE + n0 + cc;
        out[o] = out[o] * inv;
      }
    }
  }
}

extern "C" void kernel_launch(void* const* d_in, const int* in_sizes, int n_in,
                              void* d_out, int out_size, void* d_ws, size_t ws_size,
                              hipStream_t stream) {
  const int*   entRow = (const int*)d_in[0];
  const int*   entCol = (const int*)d_in[1];
  const int*   relRow = (const int*)d_in[2];
  const int*   relCol = (const int*)d_in[3];
  const float* entEmb = (const float*)d_in[4];
  const float* relEmb = (const float*)d_in[5];

  const int nEnt     = in_sizes[0];
  const int nRel     = in_sizes[2];
  const int nodeSize = in_sizes[4] / D_FEAT;   // ent_emb is [node_size, 128]

  float* out    = (float*)d_out;
  int*   degEnt = (int*)d_ws;                  // [nodeSize]
  int*   degRel = degEnt + nodeSize;           // [nodeSize]

  // 1) zero degree scratch (ws is poisoned; must re-zero every call)
  stenc_zero_f32<<<1024, 256, 0, stream>>>((float*)d_ws, (long long)2 * nodeSize);
  // 2) zero output (accumulated via atomics)
  stenc_zero_f32<<<4096, 256, 0, stream>>>(out, (long long)out_size);

  // 3) degree counts for both graphs
  {
    int total = nEnt + nRel;
    int blocks = (total + 255) / 256;
    stenc_count_deg<<<blocks, 256, 0, stream>>>(entRow, nEnt, relRow, nRel,
                                                degEnt, degRel);
  }

  // 4) edge scatters (one wave32 per edge, 8 edges per 256-thread block)
  {
    int blocks = (nEnt + 7) / 8;
    stenc_scatter_edges<<<blocks, 256, 0, stream>>>(entRow, entCol, entEmb,
                                                    out, nEnt, 0);
  }
  {
    int blocks = (nRel + 7) / 8;
    stenc_scatter_edges<<<blocks, 256, 0, stream>>>(relRow, relCol, relEmb,
                                                    out, nRel, D_FEAT);
  }

  // 5) diag(1/deg) scale via f32 WMMA, one wave per 16x16 tile
  {
    int rowTiles = (nodeSize + 15) / 16;
    int nTiles   = rowTiles * 16;              // 16 column tiles (256 cols)
    int blocks   = (nTiles * 32 + 255) / 256;
    stenc_scale_wmma<<<blocks, 256, 0, stream>>>(out, degEnt, degRel,
                                                 nodeSize, nTiles);
  }
}